// CombinedRotaryEmbedding_15418932593305
// MI455X (gfx1250) — compile-verified
//
#include <hip/hip_runtime.h>
#include <math.h>

typedef __attribute__((ext_vector_type(2))) float v2f;
typedef __attribute__((ext_vector_type(4))) float v4f;
typedef __attribute__((ext_vector_type(8))) float v8f;

#define NSTATE 1024
#define NHEAD  16
#define HDIM   64
#define NUMROT 8
#define SEQ    4096
#define BATCH  4
#define ROWS   (BATCH * SEQ * NHEAD)   // 262144 rows of 64
#define TILES  (ROWS / 16)             // 16384 16-row tiles; tile t <=> token (b,s), s = t % SEQ

// ---------------------------------------------------------------------------
// Kernel 1: compose R = (G0 G1 ... G7) @ rotation_matrix, store column-permuted
//   Rp[:, d]      = R[:, 2d]      (d = 0..31)  -> "x1" columns
//   Rp[:, 32 + d] = R[:, 2d + 1]              -> "x2" columns
// One block, 64 threads, each owns one row (Givens R@G touches only cols i,j).
// ---------------------------------------------------------------------------
__global__ void compose_R_kernel(const float* __restrict__ thetas,
                                 const float* __restrict__ pairs,
                                 const float* __restrict__ theta_scale,
                                 const float* __restrict__ rotM,
                                 float* __restrict__ Rp) {
  __shared__ float R[64 * 64];
  int r = threadIdx.x;
  if (r < 64) {
    for (int c = 0; c < 64; ++c) R[r * 64 + c] = (r == c) ? 1.0f : 0.0f;
    float ts = theta_scale[0];
    for (int k = 0; k < NUMROT; ++k) {
      int i = (int)pairs[2 * k + 0];
      int j = (int)pairs[2 * k + 1];
      float th = thetas[k] * ts;
      float c = cosf(th), s = sinf(th);
      if (i == j) {                    // degenerate Givens: only G[i,i]=c survives
        R[r * 64 + i] *= c;
      } else {                         // (R@G): col_i = c*ci + s*cj ; col_j = -s*ci + c*cj
        float a = R[r * 64 + i], b = R[r * 64 + j];
        R[r * 64 + i] =  c * a + s * b;
        R[r * 64 + j] = -s * a + c * b;
      }
    }
    // row r of (R @ rotation_matrix), written with even/odd column permutation
    for (int c2 = 0; c2 < 64; ++c2) {
      float v = 0.0f;
      for (int k = 0; k < 64; ++k) v += R[r * 64 + k] * rotM[k * 64 + c2];
      int p = (c2 & 1) ? (32 + (c2 >> 1)) : (c2 >> 1);
      Rp[r * 64 + p] = v;
    }
  }
}

// ---------------------------------------------------------------------------
// Kernel 2: per-position RoPE table, interleaved (cos, sin) per (s, d'), d'=0..31
// ---------------------------------------------------------------------------
__global__ void sincos_kernel(const float* __restrict__ inv_freq,
                              float* __restrict__ tab) {
  int idx = blockIdx.x * blockDim.x + threadIdx.x;  // over SEQ*32
  if (idx < SEQ * 32) {
    int s = idx >> 5, d = idx & 31;
    float ang = (float)s * inv_freq[d];
    tab[2 * idx + 0] = cosf(ang);
    tab[2 * idx + 1] = sinf(ang);
  }
}

// ---------------------------------------------------------------------------
// Kernel 3: streaming  out = RoPE( X(262144x64) @ Rp(64x64) )  via fp32 WMMA.
// One wave per 16-row tile (grid-stride). Rp staged in LDS, B-fragments hoisted.
// ---------------------------------------------------------------------------
__global__ void __launch_bounds__(256)
rot_rope_kernel(const float* __restrict__ X,
                const float* __restrict__ Rp,
                const float* __restrict__ tab,
                float* __restrict__ out) {
  __shared__ float lds[64 * 64];  // 16 KB of 320 KB/WGP

  int tid = threadIdx.x;
  for (int i = tid; i < 1024; i += 256)               // stage Rp (16 KB) to LDS
    ((v4f*)lds)[i] = ((const v4f*)Rp)[i];
  __syncthreads();

  int lane = tid & 31;
  int wave = tid >> 5;
  int half = lane >> 4;   // 0: lanes 0-15 (K+0,K+1), 1: lanes 16-31 (K+2,K+3)
  int l15  = lane & 15;

  // Hoist all B fragments: Bf[nb][kb] covers N=16*nb..+15, K=4*kb..+3.
  // ISA B layout (mirrors A): v0/v1 hold (k0, k0+1) with k0 = 4*kb + 2*half.
  v2f Bf[4][16];
#pragma unroll
  for (int nb = 0; nb < 4; ++nb) {
#pragma unroll
    for (int kb = 0; kb < 16; ++kb) {
      int k0 = 4 * kb + 2 * half;
      int n  = 16 * nb + l15;
      v2f b;
      b.x = lds[k0 * 64 + n];
      b.y = lds[(k0 + 1) * 64 + n];
      Bf[nb][kb] = b;
    }
  }

  int waveId = blockIdx.x * 8 + wave;     // wave-uniform => EXEC stays all-ones
  int stride = gridDim.x * 8;

  for (int t = waveId; t < TILES; t += stride) {
    const float* xrow = X + ((size_t)t * 16 + l15) * 64;
    if (t + stride < TILES)                                // next tile prefetch
      __builtin_prefetch(xrow + (size_t)stride * 16 * 64, 0, 0);

    // A fragments: lane holds row (t*16 + l15), float2 at K = 4*kb + 2*half
    v2f Af[16];
#pragma unroll
    for (int kb = 0; kb < 16; ++kb)
      Af[kb] = *(const v2f*)(xrow + 4 * kb + 2 * half);

    int s_pos = t & (SEQ - 1);            // tile = one token: s shared by all 16 rows
    float* obase = out + (size_t)t * 16 * 64;

#pragma unroll
    for (int p = 0; p < 2; ++p) {         // accA: x1 cols d'=16p..16p+15, accB: x2 same d'
      v8f accA = {}, accB = {};
#pragma unroll
      for (int kb = 0; kb < 16; ++kb) {
        accA = __builtin_amdgcn_wmma_f32_16x16x4_f32(
            false, Af[kb], false, Bf[p][kb],     (short)0, accA, false, false);
        accB = __builtin_amdgcn_wmma_f32_16x16x4_f32(
            false, Af[kb], false, Bf[p + 2][kb], (short)0, accB, false, false);
      }

      int dp = 16 * p + l15;              // d' for this lane (C/D: N = lane&15)
      v2f cs = *(const v2f*)(tab + ((size_t)s_pos * 32 + dp) * 2);
      float c = cs.x, s = cs.y;

#pragma unroll
      for (int v = 0; v < 8; ++v) {       // C/D: row M = v + 8*half
        int row = v + 8 * half;
        float a = accA[v], b = accB[v];
        obase[(size_t)row * 64 + dp]      = a * c - b * s;   // out[..., d']
        obase[(size_t)row * 64 + 32 + dp] = a * s + b * c;   // out[..., 32+d']
      }
    }
  }
}

// ---------------------------------------------------------------------------
extern "C" void kernel_launch(void* const* d_in, const int* in_sizes, int n_in,
                              void* d_out, int out_size, void* d_ws, size_t ws_size,
                              hipStream_t stream) {
  const float* x      = (const float*)d_in[0];
  const float* thetas = (const float*)d_in[1];
  const float* pairs  = (const float*)d_in[2];
  const float* tscale = (const float*)d_in[3];
  const float* rotM   = (const float*)d_in[4];
  const float* invf   = (const float*)d_in[5];
  float* out = (float*)d_out;

  float* Rp  = (float*)d_ws;          // 64*64 floats
  float* tab = Rp + 64 * 64;          // SEQ*32*2 floats (~1 MB)

  compose_R_kernel<<<1, 64, 0, stream>>>(thetas, pairs, tscale, rotM, Rp);
  sincos_kernel<<<(SEQ * 32 + 255) / 256, 256, 0, stream>>>(invf, tab);
  rot_rope_kernel<<<512, 256, 0, stream>>>(x, Rp, tab, out);
}